// MistralSparseMoE_59279138619552
// MI455X (gfx1250) — compile-verified
//
#include <hip/hip_runtime.h>
#include <hip/hip_bf16.h>

// MI455X (gfx1250) fused sparse-MoE, fp32 WMMA (V_WMMA_F32_16X16X4_F32).
// HBM-bound on weight streaming (~403MB fp32); experts processed sequentially
// so each expert's 50MB weight set stays L2-resident (192MB L2).
// GEMM main loops: double-buffered LDS slabs filled with CDNA5 async
// global->LDS copies (ASYNCcnt), overlapped with the WMMA chain.

typedef float v2f __attribute__((ext_vector_type(2)));
typedef float v8f __attribute__((ext_vector_type(8)));
typedef int v4i __attribute__((ext_vector_type(4)));

#define HDIM 1024
#define FDIM 4096
#define NEXP 8

#define KC 32      // K-chunk per pipeline stage
#define LDB 36     // LDS row stride (floats): 36*m mod 64 distinct -> no conflicts
#define SLAB (128 * LDB)

#if __has_builtin(__builtin_amdgcn_global_load_async_to_lds_b128)
#define USE_ASYNC_LD 1
#endif
#if __has_builtin(__builtin_amdgcn_global_store_async_from_lds_b128)
#define USE_ASYNC_ST 1
#endif

#if defined(USE_ASYNC_LD) || defined(USE_ASYNC_ST)
// Builtin params are int4* in AS1 (global) / AS3 (LDS). Cast through uintptr_t:
// AS1 pointers are 64-bit flat global addresses; AS3 pointers are 32-bit LDS
// offsets == low 32 bits of the generic shared address (aperture is the high
// word), so the truncating integer cast yields exactly the LDS byte offset.
typedef __attribute__((address_space(1))) v4i as1_v4i;
typedef __attribute__((address_space(3))) v4i as3_v4i;
#define AS1P(p) ((as1_v4i*)(uintptr_t)(p))
#define AS3P(p) ((as3_v4i*)(uintptr_t)(p))
#endif

__device__ __forceinline__ void copy16(const float* __restrict__ g,
                                       float* __restrict__ l) {
#if defined(USE_ASYNC_LD)
  // GLOBAL_LOAD_ASYNC_TO_LDS_B128: per-lane 16B global -> LDS, ASYNCcnt-tracked
  __builtin_amdgcn_global_load_async_to_lds_b128(AS1P(g), AS3P(l), 0, 0);
#else
  *(float4*)l = *(const float4*)g;
#endif
}

__device__ __forceinline__ void async_wait0() {
#if defined(USE_ASYNC_LD)
#if __has_builtin(__builtin_amdgcn_s_wait_asynccnt)
  __builtin_amdgcn_s_wait_asynccnt(0);
#else
  asm volatile("s_wait_asynccnt 0x0" ::: "memory");
#endif
#endif
}

__device__ __forceinline__ v8f wmma_f32(v2f a, v2f b, v8f c) {
  // D = A(16x4) x B(4x16) + C(16x16), fp32
  return __builtin_amdgcn_wmma_f32_16x16x4_f32(false, a, false, b, (short)0, c,
                                               false, false);
}

// ---------------------------------------------------------------------------
// Router: logits = x @ router_w^T  -> top-2 -> softmax -> dense combine [T,E]
// One wave (32 lanes) per token; lanes split H; shfl-xor reduction.
// ---------------------------------------------------------------------------
__global__ __launch_bounds__(256) void moe_router_kernel(
    const float* __restrict__ x, const float* __restrict__ rw,
    float* __restrict__ combine) {
  __shared__ float rws[NEXP * HDIM];  // 32KB
  const int tid = threadIdx.x;
  for (int i = tid; i < NEXP * HDIM; i += 256) rws[i] = rw[i];
  __syncthreads();

  const int wave = tid >> 5;
  const int lane = tid & 31;
  const int t = blockIdx.x * 8 + wave;

  float acc[NEXP];
#pragma unroll
  for (int e = 0; e < NEXP; ++e) acc[e] = 0.0f;

  const float* xr = x + (size_t)t * HDIM;
  for (int h = lane; h < HDIM; h += 32) {
    const float xv = xr[h];
#pragma unroll
    for (int e = 0; e < NEXP; ++e) acc[e] = fmaf(xv, rws[e * HDIM + h], acc[e]);
  }
#pragma unroll
  for (int e = 0; e < NEXP; ++e) {
    for (int off = 16; off > 0; off >>= 1) acc[e] += __shfl_xor(acc[e], off, 32);
  }

  if (lane == 0) {
    int i0 = 0;
    float v0 = acc[0];
#pragma unroll
    for (int e = 1; e < NEXP; ++e)
      if (acc[e] > v0) { v0 = acc[e]; i0 = e; }
    int i1 = (i0 == 0) ? 1 : 0;
    float v1 = acc[i1];
#pragma unroll
    for (int e = 0; e < NEXP; ++e)
      if (e != i0 && acc[e] > v1) { v1 = acc[e]; i1 = e; }
    const float e1 = __expf(v1 - v0);  // softmax over the two top logits
    const float w0 = 1.0f / (1.0f + e1);
    const float w1 = e1 / (1.0f + e1);
    float outv[NEXP];
#pragma unroll
    for (int e = 0; e < NEXP; ++e) outv[e] = 0.0f;
    outv[i0] = w0;
    outv[i1] = w1;
#pragma unroll
    for (int e = 0; e < NEXP; ++e) combine[(size_t)t * NEXP + e] = outv[e];
  }
}

// ---------------------------------------------------------------------------
// GU kernel: act[t, f] = silu(x@gw^T) * (x@uw^T) for one expert.
// 512 threads = 16 waves. 128x128 output tile: waves 0-7 -> gate tiles,
// waves 8-15 -> up tiles (shared A slab). Double-buffered K=32 slabs,
// async global->LDS fill overlapped with WMMA on the other buffer.
// ---------------------------------------------------------------------------
__global__ __launch_bounds__(512) void moe_gu_kernel(
    const float* __restrict__ x, const float* __restrict__ gw,
    const float* __restrict__ uw, float* __restrict__ act, int t_base) {
  __shared__ float As[2 * SLAB];   // 36.9 KB
  __shared__ float Bsh[4 * SLAB];  // Bg[2] | Bu[2], 73.7 KB (re-used as actbuf)

  const int tid = threadIdx.x;
  const int wave = tid >> 5;
  const int lane = tid & 31;
  const int t0 = blockIdx.x * 128;  // chunk-local token tile
  const int f0 = blockIdx.y * 128;

  const bool isG = wave < 8;
  const int mi = wave & 7;  // m-tile owned by this wave
  const int m = lane & 15;
  const int koff = (lane >> 4) * 2;

  // per-thread slab fill coordinates: 2 x b128 per slab per stage
  const int r0 = tid >> 3;          // 0..63
  const int c4 = (tid & 7) * 4;     // 0..28
  const float* gA0 = &x[(size_t)(t_base + t0 + r0) * HDIM + c4];
  const float* gA1 = &x[(size_t)(t_base + t0 + r0 + 64) * HDIM + c4];
  const float* gG0 = &gw[(size_t)(f0 + r0) * HDIM + c4];
  const float* gG1 = &gw[(size_t)(f0 + r0 + 64) * HDIM + c4];
  const float* gU0 = &uw[(size_t)(f0 + r0) * HDIM + c4];
  const float* gU1 = &uw[(size_t)(f0 + r0 + 64) * HDIM + c4];

  v8f acc[8];
#pragma unroll
  for (int j = 0; j < 8; ++j)
#pragma unroll
    for (int r = 0; r < 8; ++r) acc[j][r] = 0.0f;

#define GU_PREFETCH(buf, k0)                                      \
  do {                                                            \
    float* aA = &As[(buf)*SLAB + r0 * LDB + c4];                  \
    float* aG = &Bsh[(buf)*SLAB + r0 * LDB + c4];                 \
    float* aU = &Bsh[(2 + (buf)) * SLAB + r0 * LDB + c4];         \
    copy16(gA0 + (k0), aA);                                       \
    copy16(gA1 + (k0), aA + 64 * LDB);                            \
    copy16(gG0 + (k0), aG);                                       \
    copy16(gG1 + (k0), aG + 64 * LDB);                            \
    copy16(gU0 + (k0), aU);                                       \
    copy16(gU1 + (k0), aU + 64 * LDB);                            \
  } while (0)

  GU_PREFETCH(0, 0);
  async_wait0();
  __syncthreads();

  for (int kc = 0; kc < HDIM / KC; ++kc) {
    const int cur = kc & 1;
    if (kc + 1 < HDIM / KC) GU_PREFETCH(cur ^ 1, (kc + 1) * KC);

    const float* Ab = &As[cur * SLAB];
    const float* Bb = &Bsh[(isG ? cur : 2 + cur) * SLAB];
    for (int ks = 0; ks < KC / 4; ++ks) {
      const int k = ks * 4;
      const v2f a = *(const v2f*)&Ab[(mi * 16 + m) * LDB + k + koff];
#pragma unroll
      for (int j = 0; j < 8; ++j) {
        const v2f b = *(const v2f*)&Bb[(j * 16 + m) * LDB + k + koff];
        acc[j] = wmma_f32(a, b, acc[j]);
      }
    }
    async_wait0();   // next-buffer fill done (overlapped with the WMMAs above)
    __syncthreads(); // everyone done computing cur + filling next
  }

  float* actbuf = Bsh;  // 128 x 132 floats (67.6KB) fits in Bsh (73.7KB)
  if (isG) {
#pragma unroll
    for (int j = 0; j < 8; ++j)
#pragma unroll
      for (int r = 0; r < 8; ++r) {
        const int mm = (lane >> 4) * 8 + r;
        actbuf[(mi * 16 + mm) * 132 + j * 16 + (lane & 15)] = acc[j][r];
      }
  }
  __syncthreads();
  if (!isG) {
#pragma unroll
    for (int j = 0; j < 8; ++j)
#pragma unroll
      for (int r = 0; r < 8; ++r) {
        const int mm = (lane >> 4) * 8 + r;
        const int idx = (mi * 16 + mm) * 132 + j * 16 + (lane & 15);
        const float g = actbuf[idx];
        const float s = g * (1.0f / (1.0f + __expf(-g)));  // silu
        actbuf[idx] = s * acc[j][r];
      }
  }
  __syncthreads();
#pragma unroll
  for (int i = 0; i < 8; ++i) {
    const int idx = tid + i * 512;  // 0..4095 over [128 rows][32 float4]
    const int r = idx >> 5, c = (idx & 31) * 4;
    float* gdst = &act[(size_t)(t0 + r) * FDIM + f0 + c];
    const float* lsrc = &actbuf[r * 132 + c];
#if defined(USE_ASYNC_ST)
    // GLOBAL_STORE_ASYNC_FROM_LDS_B128; S_ENDPGM's implicit wait-idle covers it
    __builtin_amdgcn_global_store_async_from_lds_b128(AS1P(gdst), AS3P(lsrc), 0,
                                                      0);
#else
    *(float4*)gdst = *(const float4*)lsrc;
#endif
  }
#undef GU_PREFETCH
}

// ---------------------------------------------------------------------------
// Down kernel: out[t,h] (init/+=) combine[t,e] * (act @ down_w[e]^T)[t,h]
// 512 threads = 16 waves; 128x128 tile; each wave: 1 m-tile x 4 n-tiles.
// Same double-buffered async pipeline over K = F = 4096.
// ---------------------------------------------------------------------------
__global__ __launch_bounds__(512) void moe_down_kernel(
    const float* __restrict__ act, const float* __restrict__ dwexp,
    const float* __restrict__ combine, float* __restrict__ out, int t_base,
    int e, int first) {
  __shared__ float As[2 * SLAB];
  __shared__ float Bs[2 * SLAB];
  __shared__ float cmb[128];

  const int tid = threadIdx.x;
  const int wave = tid >> 5;
  const int lane = tid & 31;
  const int t0 = blockIdx.x * 128;
  const int h0 = blockIdx.y * 128;

  if (tid < 128) cmb[tid] = combine[(size_t)(t_base + t0 + tid) * NEXP + e];

  const int mi = wave >> 1;  // 0..7 m-tile
  const int nh = wave & 1;   // n half (4 tiles each)
  const int m = lane & 15;
  const int koff = (lane >> 4) * 2;

  const int r0 = tid >> 3;
  const int c4 = (tid & 7) * 4;
  const float* gA0 = &act[(size_t)(t0 + r0) * FDIM + c4];
  const float* gA1 = &act[(size_t)(t0 + r0 + 64) * FDIM + c4];
  const float* gB0 = &dwexp[(size_t)(h0 + r0) * FDIM + c4];
  const float* gB1 = &dwexp[(size_t)(h0 + r0 + 64) * FDIM + c4];

  v8f acc[4];
#pragma unroll
  for (int j = 0; j < 4; ++j)
#pragma unroll
    for (int r = 0; r < 8; ++r) acc[j][r] = 0.0f;

#define DN_PREFETCH(buf, k0)                             \
  do {                                                   \
    float* aA = &As[(buf)*SLAB + r0 * LDB + c4];         \
    float* aB = &Bs[(buf)*SLAB + r0 * LDB + c4];         \
    copy16(gA0 + (k0), aA);                              \
    copy16(gA1 + (k0), aA + 64 * LDB);                   \
    copy16(gB0 + (k0), aB);                              \
    copy16(gB1 + (k0), aB + 64 * LDB);                   \
  } while (0)

  DN_PREFETCH(0, 0);
  async_wait0();
  __syncthreads();

  for (int kc = 0; kc < FDIM / KC; ++kc) {
    const int cur = kc & 1;
    if (kc + 1 < FDIM / KC) DN_PREFETCH(cur ^ 1, (kc + 1) * KC);

    const float* Ab = &As[cur * SLAB];
    const float* Bb = &Bs[cur * SLAB];
    for (int ks = 0; ks < KC / 4; ++ks) {
      const int k = ks * 4;
      const v2f a = *(const v2f*)&Ab[(mi * 16 + m) * LDB + k + koff];
#pragma unroll
      for (int j = 0; j < 4; ++j) {
        const v2f b = *(const v2f*)&Bb[((nh * 4 + j) * 16 + m) * LDB + k + koff];
        acc[j] = wmma_f32(a, b, acc[j]);
      }
    }
    async_wait0();
    __syncthreads();
  }

#pragma unroll
  for (int j = 0; j < 4; ++j) {
#pragma unroll
    for (int r = 0; r < 8; ++r) {
      const int mm = (lane >> 4) * 8 + r;
      const float c = cmb[mi * 16 + mm];
      const size_t o = (size_t)(t_base + t0 + mi * 16 + mm) * HDIM + h0 +
                       (nh * 4 + j) * 16 + (lane & 15);
      const float v = acc[j][r] * c;
      if (first) out[o] = v;
      else out[o] += v;
    }
  }
#undef DN_PREFETCH
}

// ---------------------------------------------------------------------------
extern "C" void kernel_launch(void* const* d_in, const int* in_sizes, int n_in,
                              void* d_out, int out_size, void* d_ws,
                              size_t ws_size, hipStream_t stream) {
  const float* x = (const float*)d_in[0];   // [B,S,H]
  const float* rw = (const float*)d_in[1];  // [E,H]
  const float* gw = (const float*)d_in[2];  // [E,F,H]
  const float* uw = (const float*)d_in[3];  // [E,F,H]
  const float* dw = (const float*)d_in[4];  // [E,H,F]
  float* out = (float*)d_out;

  const int T = in_sizes[0] / HDIM;  // 8192 tokens

  float* combine = (float*)d_ws;
  const size_t comb_bytes = ((size_t)T * NEXP * sizeof(float) + 255) & ~255ull;
  float* act = (float*)((char*)d_ws + comb_bytes);
  const size_t avail = (ws_size > comb_bytes) ? (ws_size - comb_bytes) : 0;

  long long tc = (long long)(avail / ((size_t)FDIM * sizeof(float)));
  tc = (tc / 128) * 128;
  if (tc < 128) tc = 128;
  if (tc > T) tc = T;

  moe_router_kernel<<<T / 8, 256, 0, stream>>>(x, rw, combine);

  for (int t_base = 0; t_base < T; t_base += (int)tc) {
    int rows = T - t_base;
    if (rows > tc) rows = (int)tc;
    dim3 gGU(rows / 128, FDIM / 128);
    dim3 gD(rows / 128, HDIM / 128);
    for (int e = 0; e < NEXP; ++e) {
      moe_gu_kernel<<<gGU, 512, 0, stream>>>(
          x, gw + (size_t)e * FDIM * HDIM, uw + (size_t)e * FDIM * HDIM, act,
          t_base);
      moe_down_kernel<<<gD, 512, 0, stream>>>(
          act, dw + (size_t)e * HDIM * FDIM, combine, out, t_base, e,
          e == 0 ? 1 : 0);
    }
  }
}